// FrigateYOLOModel_11141145166152
// MI455X (gfx1250) — compile-verified
//
#include <hip/hip_runtime.h>
#include <hip/hip_bf16.h>

// -----------------------------------------------------------------------------
// FrigateYOLOModel post-processing (per-anchor class max + box decode + NMS)
// MI455X / gfx1250 design:
//  * 1 workgroup per image (64 WGs), 1024 threads = 32 wave32 waves.
//  * Entire per-image NMS state (boxes as float4 AoS + scores + classes,
//    ~202 KB) resident in CDNA5's 320 KB WGP LDS; all 100 serial NMS
//    iterations sweep LDS with b128 accesses (1 ds_load_b128 per anchor).
//  * async global->LDS b128 DMA (ASYNCcnt) stages the 4 raw box rows while
//    the 32-channel class-max streams from global with float4 loads.
//  * wave32 shuffle tree (ds_bpermute/permlane) for the argmax reduction.
//  * global_prefetch_b8 software pipelining of the strided class rows.
// -----------------------------------------------------------------------------

#define NUM_CLASSES 32
#define MAX_DET     100
#define CONF_T      0.25f
#define IOU_T       0.45f
#define BATCH       64
#define ANCHORS     8400
#define NQ          (ANCHORS / 4)        // 2100 float4 groups
#define NCH         (4 + NUM_CLASSES)    // 36
#define THREADS     1024
#define NWAVES      (THREADS / 32)       // 32
#define MAXQ        3                    // ceil(NQ / THREADS)

// LDS: s_box float4[ANCHORS] | s_score float[ANCHORS] | s_cls float[ANCHORS]
//      | red_val[32] | red_idx[32] | bc_val | bc_idx
#define SMEM_BYTES  ((size_t)ANCHORS * 16 + (size_t)ANCHORS * 8 + NWAVES * 8 + 16)

typedef int v4i __attribute__((ext_vector_type(4)));

#if defined(__has_builtin)
#  if __has_builtin(__builtin_amdgcn_global_load_async_to_lds_b32) && \
      __has_builtin(__builtin_amdgcn_s_wait_asynccnt)
#    define USE_ASYNC 1
#  endif
#  if defined(USE_ASYNC) && __has_builtin(__builtin_amdgcn_global_load_async_to_lds_b128)
#    define USE_ASYNC_B128 1
#  endif
#endif
#ifndef USE_ASYNC
#  define USE_ASYNC 0
#endif
#ifndef USE_ASYNC_B128
#  define USE_ASYNC_B128 0
#endif

#if USE_ASYNC
typedef __attribute__((address_space(1))) int g_i32;
typedef __attribute__((address_space(3))) int l_i32;
typedef __attribute__((address_space(1))) v4i g_v4i;
typedef __attribute__((address_space(3))) v4i l_v4i;
#endif

// Copy 16 aligned bytes global -> LDS (async DMA when available).
__device__ __forceinline__ void async_copy16(const float* g, float* l) {
#if USE_ASYNC_B128
    __builtin_amdgcn_global_load_async_to_lds_b128((g_v4i*)g, (l_v4i*)l, 0, 0);
#elif USE_ASYNC
    __builtin_amdgcn_global_load_async_to_lds_b32((g_i32*)(g + 0), (l_i32*)(l + 0), 0, 0);
    __builtin_amdgcn_global_load_async_to_lds_b32((g_i32*)(g + 1), (l_i32*)(l + 1), 0, 0);
    __builtin_amdgcn_global_load_async_to_lds_b32((g_i32*)(g + 2), (l_i32*)(l + 2), 0, 0);
    __builtin_amdgcn_global_load_async_to_lds_b32((g_i32*)(g + 3), (l_i32*)(l + 3), 0, 0);
#else
    *(float4*)l = *(const float4*)g;
#endif
}

__device__ __forceinline__ void wait_async_all() {
#if USE_ASYNC
    __builtin_amdgcn_s_wait_asynccnt(0);
#endif
}

// IoU-suppress one score against the selected box B (layout y1,x1,y2,x2).
__device__ __forceinline__ float supp1(float s, float4 bb, float4 B, float areaA, bool isSel) {
    float iy1 = fmaxf(B.x, bb.x);
    float ix1 = fmaxf(B.y, bb.y);
    float iy2 = fminf(B.z, bb.z);
    float ix2 = fminf(B.w, bb.w);
    float inter = fmaxf(iy2 - iy1, 0.0f) * fmaxf(ix2 - ix1, 0.0f);
    float areaB = (bb.z - bb.x) * (bb.w - bb.y);
    float uni   = areaA + areaB - inter;
    float iou   = (uni > 0.0f) ? (inter / uni) : 0.0f;
    return (iou > IOU_T || isSel) ? -1.0f : s;
}

__global__ __launch_bounds__(THREADS, 1)
void yolo_nms_kernel(const float* __restrict__ in,   // (B, 36, 8400)
                     float* __restrict__ det_boxes,  // (B, 100, 4)
                     float* __restrict__ det_cls,    // (B, 100)
                     float* __restrict__ det_sc,     // (B, 100)
                     float* __restrict__ ndet_out)   // (B,)
{
    extern __shared__ __align__(16) char smem_raw[];
    float4* s_box    = (float4*)smem_raw;                     // [ANCHORS] (y1,x1,y2,x2)
    float*  raw      = (float*)smem_raw;                      // phase-1 SoA staging alias
    float*  s_score  = (float*)(smem_raw + (size_t)ANCHORS * 16); // [ANCHORS]
    float*  s_cls    = s_score + ANCHORS;                     // [ANCHORS]
    float*  red_val  = s_cls + ANCHORS;                       // [NWAVES]
    int*    red_idx  = (int*)(red_val + NWAVES);              // [NWAVES]
    float*  bc_val   = (float*)(red_idx + NWAVES);
    int*    bc_idx   = (int*)(bc_val + 1);
    float4* s_score4 = (float4*)s_score;
    float4* s_cls4   = (float4*)s_cls;

    const int b   = blockIdx.x;
    const int tid = threadIdx.x;
    const float* __restrict__ img = in + (size_t)b * NCH * ANCHORS;

    // ---- Phase 1a: async-DMA the 4 raw box rows (SoA) into the box region ---
    #pragma unroll
    for (int i = 0; i < MAXQ; ++i) {
        int q = tid + i * THREADS;
        if (q < NQ) {
            #pragma unroll
            for (int c = 0; c < 4; ++c)
                async_copy16(img + (size_t)c * ANCHORS + 4 * q,
                             raw + (size_t)c * ANCHORS + 4 * q);
        }
    }

    // ---- Phase 1b: class max/argmax from global (overlaps DMA) -------------
    #pragma unroll
    for (int i = 0; i < MAXQ; ++i) {
        int q = tid + i * THREADS;
        if (q < NQ) {
            if (q + THREADS < NQ) {  // prefetch next chunk's class rows
                #pragma unroll
                for (int c = 4; c < NCH; ++c)
                    __builtin_prefetch(img + (size_t)c * ANCHORS + 4 * (q + THREADS), 0, 1);
            }
            float4 best = ((const float4*)(img + (size_t)4 * ANCHORS))[q];
            float4 cls  = make_float4(0.0f, 0.0f, 0.0f, 0.0f);
            #pragma unroll
            for (int c = 1; c < NUM_CLASSES; ++c) {
                float4 v = ((const float4*)(img + (size_t)(4 + c) * ANCHORS))[q];
                if (v.x > best.x) { best.x = v.x; cls.x = (float)c; }
                if (v.y > best.y) { best.y = v.y; cls.y = (float)c; }
                if (v.z > best.z) { best.z = v.z; cls.z = (float)c; }
                if (v.w > best.w) { best.w = v.w; cls.w = (float)c; }
            }
            float4 sc;
            sc.x = (best.x >= CONF_T) ? best.x : -1.0f;
            sc.y = (best.y >= CONF_T) ? best.y : -1.0f;
            sc.z = (best.z >= CONF_T) ? best.z : -1.0f;
            sc.w = (best.w >= CONF_T) ? best.w : -1.0f;
            s_score4[q] = sc;
            s_cls4[q]   = cls;
        }
    }

    // ---- Phase 1c: in-place SoA -> AoS corner transform --------------------
    wait_async_all();                 // each thread reads only its own DMAs
    float4 rxc[MAXQ], ryc[MAXQ], rww[MAXQ], rhh[MAXQ];
    const float4* raw4 = (const float4*)raw;
    #pragma unroll
    for (int i = 0; i < MAXQ; ++i) {
        int q = tid + i * THREADS;
        if (q < NQ) {
            rxc[i] = raw4[0 * NQ + q];
            ryc[i] = raw4[1 * NQ + q];
            rww[i] = raw4[2 * NQ + q];
            rhh[i] = raw4[3 * NQ + q];
        }
    }
    __syncthreads();                  // all raw reads done before AoS writes
    #pragma unroll
    for (int i = 0; i < MAXQ; ++i) {
        int q = tid + i * THREADS;
        if (q < NQ) {
            float xs[4] = {rxc[i].x, rxc[i].y, rxc[i].z, rxc[i].w};
            float ys[4] = {ryc[i].x, ryc[i].y, ryc[i].z, ryc[i].w};
            float ws[4] = {rww[i].x, rww[i].y, rww[i].z, rww[i].w};
            float hs[4] = {rhh[i].x, rhh[i].y, rhh[i].z, rhh[i].w};
            #pragma unroll
            for (int k = 0; k < 4; ++k) {
                float y1 = fminf(fmaxf(ys[k] - hs[k] * 0.5f, 0.0f), 1.0f);
                float x1 = fminf(fmaxf(xs[k] - ws[k] * 0.5f, 0.0f), 1.0f);
                float y2 = fminf(fmaxf(ys[k] + hs[k] * 0.5f, 0.0f), 1.0f);
                float x2 = fminf(fmaxf(xs[k] + ws[k] * 0.5f, 0.0f), 1.0f);
                s_box[4 * q + k] = make_float4(y1, x1, y2, x2);   // ds_store_b128
            }
        }
    }
    __syncthreads();

    // ---- Phase 2: sequential NMS, all state in LDS -------------------------
    bool done = false;
    int  ndet = 0;

    for (int it = 0; it < MAX_DET; ++it) {
        if (done) {                      // uniform across the whole block
            if (tid == 0) {
                size_t bo = ((size_t)b * MAX_DET + it) * 4;
                det_boxes[bo + 0] = 0.0f;
                det_boxes[bo + 1] = 0.0f;
                det_boxes[bo + 2] = 0.0f;
                det_boxes[bo + 3] = 0.0f;
                det_cls[b * MAX_DET + it] = 0.0f;
                det_sc [b * MAX_DET + it] = 0.0f;
            }
            continue;
        }

        // --- block-wide argmax (first-index tie-break), float4 sweeps ---
        float bv = -__builtin_huge_valf();
        int   bi = 0x7fffffff;
        for (int q = tid; q < NQ; q += THREADS) {
            float4 s = s_score4[q];      // ds_load_b128
            int a = 4 * q;
            if (s.x > bv) { bv = s.x; bi = a; }
            if (s.y > bv) { bv = s.y; bi = a + 1; }
            if (s.z > bv) { bv = s.z; bi = a + 2; }
            if (s.w > bv) { bv = s.w; bi = a + 3; }
        }
        #pragma unroll
        for (int off = 16; off > 0; off >>= 1) {
            float ov = __shfl_down(bv, off, 32);
            int   oi = __shfl_down(bi, off, 32);
            if (ov > bv || (ov == bv && oi < bi)) { bv = ov; bi = oi; }
        }
        if ((tid & 31) == 0) { red_val[tid >> 5] = bv; red_idx[tid >> 5] = bi; }
        __syncthreads();
        if (tid < NWAVES) {              // wave 0 combines the 32 wave results
            bv = red_val[tid];
            bi = red_idx[tid];
            #pragma unroll
            for (int off = 16; off > 0; off >>= 1) {
                float ov = __shfl_down(bv, off, 32);
                int   oi = __shfl_down(bi, off, 32);
                if (ov > bv || (ov == bv && oi < bi)) { bv = ov; bi = oi; }
            }
            if (tid == 0) { *bc_val = bv; *bc_idx = bi; }
        }
        __syncthreads();

        float sv = *bc_val;
        int   si = *bc_idx;
        bool  ok = sv > 0.0f;
        float4 B = s_box[si];

        if (tid == 0) {
            size_t bo = ((size_t)b * MAX_DET + it) * 4;
            det_boxes[bo + 0] = ok ? B.x : 0.0f;
            det_boxes[bo + 1] = ok ? B.y : 0.0f;
            det_boxes[bo + 2] = ok ? B.z : 0.0f;
            det_boxes[bo + 3] = ok ? B.w : 0.0f;
            det_cls[b * MAX_DET + it] = ok ? s_cls[si] : 0.0f;
            det_sc [b * MAX_DET + it] = ok ? sv : 0.0f;
            if (ok) ++ndet;
        }

        if (ok) {
            float areaA = (B.z - B.x) * (B.w - B.y);
            for (int q = tid; q < NQ; q += THREADS) {
                float4 sc = s_score4[q];
                int a = 4 * q;
                float4 b0 = s_box[a + 0];     // ds_load_b128 x4
                float4 b1 = s_box[a + 1];
                float4 b2 = s_box[a + 2];
                float4 b3 = s_box[a + 3];
                sc.x = supp1(sc.x, b0, B, areaA, (a + 0) == si);  // includes .at[idx].set(-1)
                sc.y = supp1(sc.y, b1, B, areaA, (a + 1) == si);
                sc.z = supp1(sc.z, b2, B, areaA, (a + 2) == si);
                sc.w = supp1(sc.w, b3, B, areaA, (a + 3) == si);
                s_score4[q] = sc;             // ds_store_b128
            }
        } else {
            done = true;   // scores only ever decrease; all later iters empty
        }
        __syncthreads();
    }

    if (tid == 0) ndet_out[b] = (float)ndet;
}

extern "C" void kernel_launch(void* const* d_in, const int* in_sizes, int n_in,
                              void* d_out, int out_size, void* d_ws, size_t ws_size,
                              hipStream_t stream) {
    (void)in_sizes; (void)n_in; (void)out_size; (void)d_ws; (void)ws_size;

    const float* in  = (const float*)d_in[0];
    float*       out = (float*)d_out;

    // Output layout: boxes (64*100*4) | classes (64*100) | scores (64*100) | num_det (64)
    float* det_boxes = out;
    float* det_cls   = det_boxes + (size_t)BATCH * MAX_DET * 4;
    float* det_sc    = det_cls   + (size_t)BATCH * MAX_DET;
    float* ndet      = det_sc    + (size_t)BATCH * MAX_DET;

    // ~202 KB dynamic LDS per workgroup (CDNA5 WGP has 320 KB); raise the cap.
    // Deterministic, not a stream op -> safe under graph capture.
    (void)hipFuncSetAttribute((const void*)yolo_nms_kernel,
                              hipFuncAttributeMaxDynamicSharedMemorySize,
                              (int)SMEM_BYTES);

    yolo_nms_kernel<<<BATCH, THREADS, SMEM_BYTES, stream>>>(in, det_boxes, det_cls, det_sc, ndet);
}